// ColorHistogramLoss_51582557225748
// MI455X (gfx1250) — compile-verified
//
#include <hip/hip_runtime.h>
#include <hip/hip_bf16.h>

typedef float v4f __attribute__((ext_vector_type(4)));

#define HW          262144      // 512*512 pixels per plane
#define GPI         65536       // HW/4 float4-groups per image
#define IMG_STRIDE  786432      // 3*HW floats per image
#define LSTR        61          // odd stride -> 32 lanes hit 32 distinct LDS banks

// Per-pixel HSV -> three bin increments into this lane's private LDS region.
// hh[0..9]=hue hist, hh[10..19]=sat hist, hh[20..29]=val hist.
__device__ __forceinline__ void accum_px(float r, float g, float b, unsigned* hh) {
    float mx = fmaxf(r, fmaxf(g, b));          // v_max3
    float mn = fminf(r, fminf(g, b));          // v_min3
    float d  = mx - mn;
    bool  dz = (d == 0.0f);
    float rd = __builtin_amdgcn_rcpf(dz ? 1.0f : d);   // v_rcp_f32, ~1ulp
    // hue in [0,6): branchless replica of the reference's nested wheres
    float t1 = (g - b) * rd;                   // mx==r arm (needs mod 6)
    t1 = (t1 < 0.0f) ? (t1 + 6.0f) : t1;
    float t2 = __builtin_fmaf(b - r, rd, 2.0f);  // mx==g arm
    float t3 = __builtin_fmaf(r - g, rd, 4.0f);  // mx==b arm (highest priority)
    float hue6 = (mx == g) ? t2 : t1;
    hue6 = (mx == b) ? t3 : hue6;
    hue6 = dz ? 0.0f : hue6;
    float rmx = __builtin_amdgcn_rcpf((mx == 0.0f) ? 1.0f : mx);
    // bins = clamp(floor(x*10), 0, 9); float clamp compiles to v_med3
    int bh = (int)fminf(fmaxf(hue6 * 1.66666667f, 0.0f), 9.0f);   // 10/6
    int bs = (int)fminf(fmaxf((d * 10.0f) * rmx, 0.0f), 9.0f);
    int bv = (int)fminf(fmaxf(mx * 10.0f,        0.0f), 9.0f);
    atomicAdd(&hh[bh], 1u);        // ds_add_u32 (no return), conflict-free
    atomicAdd(&hh[10 + bs], 1u);
    atomicAdd(&hh[20 + bv], 1u);
}

__global__ __launch_bounds__(256)
void ColorHistogramLoss_hist6(const float* __restrict__ xr,
                              const float* __restrict__ xf,
                              unsigned* __restrict__ gh, int ngroups) {
    __shared__ unsigned H[256 * LSTR];   // per-lane private histograms (62.5 KB)
    __shared__ unsigned P[256];          // chunk partials for reduction
    const int t = threadIdx.x;
    unsigned* h = &H[t * LSTR];
#pragma unroll
    for (int i = 0; i < 60; ++i) h[i] = 0;
    __syncthreads();

    const int stride = gridDim.x * blockDim.x;
    for (int g = blockIdx.x * blockDim.x + t; g < ngroups; g += stride) {
        const int n = g >> 16;                    // image index (GPI == 1<<16)
        const int o = g & (GPI - 1);              // float4-group within plane
        const size_t base = (size_t)n * IMG_STRIDE + (size_t)o * 4;
        // streaming 128-bit non-temporal loads: data is touched exactly once
        v4f Rr = __builtin_nontemporal_load((const v4f*)(xr + base));
        v4f Gr = __builtin_nontemporal_load((const v4f*)(xr + base + HW));
        v4f Br = __builtin_nontemporal_load((const v4f*)(xr + base + 2 * HW));
        v4f Rf = __builtin_nontemporal_load((const v4f*)(xf + base));
        v4f Gf = __builtin_nontemporal_load((const v4f*)(xf + base + HW));
        v4f Bf = __builtin_nontemporal_load((const v4f*)(xf + base + 2 * HW));
#pragma unroll
        for (int k = 0; k < 4; ++k) {
            accum_px(Rr[k], Gr[k], Br[k], h);        // real -> bins 0..29
            accum_px(Rf[k], Gf[k], Bf[k], h + 30);   // fake -> bins 30..59
        }
    }
    __syncthreads();

    // Reduce 256 lane-private copies -> 60 block totals -> global atomics.
    const int b = t & 63;            // bin id (0..59 valid)
    const int c = t >> 6;            // 64-lane chunk id (0..3)
    unsigned s = 0;
    if (b < 60) {
        const unsigned* col = &H[(c * 64) * LSTR + b];
        for (int i = 0; i < 64; ++i) s += col[i * LSTR];
    }
    P[t] = s;
    __syncthreads();
    if (t < 60) {
        unsigned tot = P[t] + P[t + 64] + P[t + 128] + P[t + 192];
        atomicAdd(&gh[t], tot);      // only 60 global atomics per block
    }
}

__global__ void ColorHistogramLoss_zero(unsigned* gh) {
    if (threadIdx.x < 64) gh[threadIdx.x] = 0u;
}

__global__ void ColorHistogramLoss_finish(const unsigned* __restrict__ gh,
                                          float* __restrict__ out) {
    if (threadIdx.x == 0) {
        // counts <= 8.39M are exactly representable in fp32 (< 2^24)
        const float w[3] = {0.3f, 0.4f, 0.4f};
        float loss = 0.0f;
        for (int ch = 0; ch < 3; ++ch) {
            float s = 0.0f;
            for (int b = 0; b < 10; ++b) {
                float dr = (float)gh[ch * 10 + b];
                float df = (float)gh[30 + ch * 10 + b];
                s += fabsf(dr - df);
            }
            loss += w[ch] * (s * 0.1f);   // mean over 10 bins
        }
        out[0] = loss;
    }
}

extern "C" void kernel_launch(void* const* d_in, const int* in_sizes, int n_in,
                              void* d_out, int out_size, void* d_ws, size_t ws_size,
                              hipStream_t stream) {
    const float* xr = (const float*)d_in[0];
    const float* xf = (const float*)d_in[1];
    unsigned* gh = (unsigned*)d_ws;          // 64 u32 counters (256 B of scratch)
    float* out = (float*)d_out;

    const int npix    = in_sizes[0] / 3;     // 8,388,608
    const int ngroups = npix / 4;            // 2,097,152 float4-groups per tensor

    ColorHistogramLoss_zero<<<1, 64, 0, stream>>>(gh);
    ColorHistogramLoss_hist6<<<512, 256, 0, stream>>>(xr, xf, gh, ngroups);
    ColorHistogramLoss_finish<<<1, 32, 0, stream>>>(gh, out);
}